// LocalAutoCorr2D_51007031607353
// MI455X (gfx1250) — compile-verified
//
#include <hip/hip_runtime.h>

// LocalAutoCorr2D on MI455X (gfx1250):
//   out[b,h,w,c,dy,dx] = sum_{u,v} x[b,c,h+u-1,w+v-1] * x[b,c,h+u-1+sy,w+v-1+sx]
//   (sy=dy-1, sx=dx-1, zero padded).
// Per shift s: P_s(i,j)=x(i,j)*x(i+sy,j+sx); out = 3x3 box-sum of P_s.
// Horizontal 3-tap on VALU from register-staged operands; vertical 3-tap as
// fp32 GEMM with a banded 0/1 matrix via V_WMMA_F32_16X16X4_F32 (fp32-exact).
// Memory roofline ~95MB @ 23.3TB/s ~ 4us; this revision cuts VALU/DS issue
// (hoisted band A, register-staged factors, ds_store_2addr-friendly row map)
// so instruction issue no longer rivals the memory floor.

typedef float v2f __attribute__((ext_vector_type(2)));
typedef float v8f __attribute__((ext_vector_type(8)));

#define Bn 4
#define Cn 64
#define Hn 96
#define Wn 96
#define TILE 16
#define THT  6   // 96/16 tiles per axis

__global__ __launch_bounds__(32)
void lac2d_wmma_kernel(const float* __restrict__ x, float* __restrict__ out) {
    // x tile with halo: rows/cols (h0-2 .. h0+17) -> 20x20
    __shared__ float xs[20][20];
    // T rows i=0..17 (global h0-1+i), cols 0..15; rows 18,19 zero (K pad 18->20)
    __shared__ float Ts[20][16];

    const int lane = threadIdx.x;            // 0..31, one wave
    const int tix  = blockIdx.x;             // 0..35
    const int c    = blockIdx.y;
    const int b    = blockIdx.z;
    const int h0   = (tix / THT) * TILE;
    const int w0   = (tix % THT) * TILE;

    const float* xp = x + (size_t)(b * Cn + c) * (Hn * Wn);

    // ---- load 20x20 halo tile, zero-padded at image borders ----
    #pragma unroll
    for (int rep = 0; rep < 13; ++rep) {     // 13*32 >= 400
        int idx = lane + rep * 32;
        if (idx < 400) {
            int r = idx / 20, q = idx % 20;
            int gh = h0 - 2 + r, gw = w0 - 2 + q;
            float v = 0.0f;
            if (gh >= 0 && gh < Hn && gw >= 0 && gw < Wn)
                v = xp[gh * Wn + gw];
            xs[r][q] = v;
        }
    }
    Ts[18 + (lane >> 4)][lane & 15] = 0.0f;  // K-padding rows, written once
    __syncthreads();

    const int w     = lane & 15;  // column owned by this lane (T, B-N, D-N)
    const int hi    = lane >> 4;  // half-wave selector
    const int rbase = 9 * hi;     // T rows computed by this lane: rbase..rbase+8
                                  // (consecutive -> ds_store_2addr pairing)

    // ---- hoisted band matrix A: Sv[M,K] = (M <= K <= M+2), M = lane&15 ----
    // A 16x4 f32 layout: vgpr0 = A[M, 4k+2*hi], vgpr1 = A[M, 4k+2*hi+1]
    v2f aband[5];
    #pragma unroll
    for (int k = 0; k < 5; ++k) {
        const int k0 = 4 * k + 2 * hi;
        aband[k].x = (k0     >= w && k0     <= w + 2) ? 1.0f : 0.0f;
        aband[k].y = (k0 + 1 >= w && k0 + 1 <= w + 2) ? 1.0f : 0.0f;
    }

    // ---- unshifted factor, staged in registers once (shift-invariant) ----
    // u[rr][vv] = x(h0-1 + rbase+rr, w0+w-1 + vv) = xs[rbase+rr+1][w+1+vv]
    float u[9][3];
    #pragma unroll
    for (int rr = 0; rr < 9; ++rr)
        #pragma unroll
        for (int vv = 0; vv < 3; ++vv)
            u[rr][vv] = xs[rbase + rr + 1][w + 1 + vv];

    v8f acc[9];

    #pragma unroll
    for (int syi = 0; syi < 3; ++syi) {
        const int sy = syi - 1;

        // shifted rows for this sy: 5-col sliding window shared by sx=-1,0,1
        // sh[rr][j] = xs[rbase+rr+1+sy][w+j], rows in [0,19], cols in [0,19]
        float sh[9][5];
        #pragma unroll
        for (int rr = 0; rr < 9; ++rr)
            #pragma unroll
            for (int j = 0; j < 5; ++j)
                sh[rr][j] = xs[rbase + rr + 1 + sy][w + j];

        #pragma unroll
        for (int sxi = 0; sxi < 3; ++sxi) {
            const int s = syi * 3 + sxi;   // = dy*3 + dx

            // T(i,w) = sum_vv u[rr][vv] * sh[rr][sxi+vv]  (all registers)
            #pragma unroll
            for (int rr = 0; rr < 9; ++rr) {
                float t =  u[rr][0] * sh[rr][sxi];
                t       += u[rr][1] * sh[rr][sxi + 1];
                t       += u[rr][2] * sh[rr][sxi + 2];
                Ts[rbase + rr][w] = t;
            }
            __syncthreads();

            // vertical 3-tap: Out = Sv(16x20) @ T(20x16), 5 fp32 WMMAs
            // B 4x16 layout: vgpr0 = B[4k+2*hi, N], vgpr1 = B[4k+2*hi+1, N]
            v8f d = {};
            #pragma unroll
            for (int k = 0; k < 5; ++k) {
                const int k0 = 4 * k + 2 * hi;
                v2f bb;
                bb.x = Ts[k0][w];
                bb.y = Ts[k0 + 1][w];
                d = __builtin_amdgcn_wmma_f32_16x16x4_f32(
                        /*neg_a=*/false, aband[k], /*neg_b=*/false, bb,
                        /*c_mod=*/(short)0, d, /*reuse_a=*/false, /*reuse_b=*/false);
            }
            acc[s] = d;
            __syncthreads();   // Ts overwritten by next shift
        }
    }

    // ---- store: D vgpr r -> pixel (h0 + r + 8*hi, w0 + (lane&15)) ----
    // 9 shift results contiguous (c,dy,dx fastest) -> b128+b128+b32 per pixel;
    // whole 85MB output fits in 192MB L2, lines written back to HBM once.
    #pragma unroll
    for (int r = 0; r < 8; ++r) {
        const int gh = h0 + r + 8 * hi;
        const int gw = w0 + w;
        float* o = out + ((size_t)(b * Hn + gh) * Wn + gw) * (size_t)(Cn * 9)
                       + c * 9;
        #pragma unroll
        for (int s = 0; s < 9; ++s)
            o[s] = acc[s][r];
    }
}

extern "C" void kernel_launch(void* const* d_in, const int* in_sizes, int n_in,
                              void* d_out, int out_size, void* d_ws, size_t ws_size,
                              hipStream_t stream) {
    (void)in_sizes; (void)n_in; (void)d_ws; (void)ws_size; (void)out_size;
    const float* x = (const float*)d_in[0];
    float* out = (float*)d_out;
    dim3 grid(THT * THT, Cn, Bn);   // (36 tiles, 64 channels, 4 batch)
    dim3 block(32);                 // one wave32 per tile -> EXEC all-ones for WMMA
    lac2d_wmma_kernel<<<grid, block, 0, stream>>>(x, out);
}